// dHOTRG_31722628448705
// MI455X (gfx1250) — compile-verified
//
#include <hip/hip_runtime.h>
#include <math.h>

// ---------------------------------------------------------------------------
// HOTRG (2D Ising) on MI455X / gfx1250.
// All heavy math is f32 GEMM -> V_WMMA_F32_16X16X4_F32 (full precision).
//   - expm via scaling (2^-8) + 12-term Taylor Horner + 8 squarings (GEMMs)
//   - the three einsum contractions as GEMMs with fused permuted epilogues
//   - tile staging via Tensor Data Mover (tensor_load_to_lds), double-buffered
//     LDS, s_wait_tensorcnt pipelining; manual fallback for non-contiguous A
//   - lam via atomicMax on uint bits of |x| (deterministic)
// ---------------------------------------------------------------------------

typedef __attribute__((ext_vector_type(2))) float v2f;
typedef __attribute__((ext_vector_type(8))) float v8f;

#define TBM 64
#define TBN 64
#define TBK 16

#if defined(__gfx1250__) && __has_builtin(__builtin_amdgcn_tensor_load_to_lds)
#define HAVE_TDM 1
#else
#define HAVE_TDM 0
#endif

#if HAVE_TDM
typedef __attribute__((ext_vector_type(4))) unsigned tdm_v4u;
typedef __attribute__((ext_vector_type(8))) int      tdm_v8i;
typedef __attribute__((ext_vector_type(4))) int      tdm_v4i;

// 2D tile load via TDM. Tile = tile_rows x tile_cols elements (f32), dim0
// (cols) contiguous in memory with row stride stride_elems. tensor_dim0/1 are
// the remaining valid cols/rows from the tile origin: TDM zero-fills OOB
// reads, giving the zero-padding the GEMM tiles require. LDS padding between
// rows via pad_interval/pad_amount (encoded per D# group1 spec).
__device__ __forceinline__ void tdm_tile_load(
    unsigned lds_byte, const float* g, int rows_rem, int cols_rem,
    unsigned tile_cols, unsigned tile_rows, unsigned stride_elems,
    unsigned pad_interval, unsigned pad_amount)
{
    unsigned long long ga = (unsigned long long)(uintptr_t)g;
    tdm_v4u g0;
    g0[0] = 1u;                                   // count=1, user mode
    g0[1] = lds_byte;                             // lds_addr
    g0[2] = (unsigned)ga;                         // global_addr[31:0]
    g0[3] = (unsigned)((ga >> 32) & 0x1FFFFFFull) // global_addr[56:32]
          | (2u << 30);                           // type = 2 (image)
    unsigned td0 = cols_rem > 0 ? (unsigned)cols_rem : 0u;
    unsigned td1 = rows_rem > 0 ? (unsigned)rows_rem : 0u;
    tdm_v8i g1;
    g1[0] = (int)((2u << 16)                      // data_size = 4 bytes
          | (1u << 20)                            // pad_enable
          | (pad_interval << 22) | (pad_amount << 25));
    g1[1] = (int)((td0 & 0xFFFFu) << 16);         // tensor_dim0[15:0]
    g1[2] = (int)((td0 >> 16) | ((td1 & 0xFFFFu) << 16));
    g1[3] = (int)((td1 >> 16) | (tile_cols << 16));
    g1[4] = (int)tile_rows;                       // tile_dim1 (tile_dim2 = 0)
    g1[5] = (int)stride_elems;                    // tensor_dim0_stride[31:0]
    g1[6] = 0;
    g1[7] = 0;
    tdm_v4i z4 = {0, 0, 0, 0};
    tdm_v8i z8 = {0, 0, 0, 0, 0, 0, 0, 0};
    __builtin_amdgcn_tensor_load_to_lds(g0, g1, z4, z4, z8, 0);
}

__device__ __forceinline__ void tdm_wait0() {
#if __has_builtin(__builtin_amdgcn_s_wait_tensorcnt)
    __builtin_amdgcn_s_wait_tensorcnt(0);
#else
    asm volatile("s_wait_tensorcnt 0x0" ::: "memory");
#endif
}
__device__ __forceinline__ void tdm_wait2() {
#if __has_builtin(__builtin_amdgcn_s_wait_tensorcnt)
    __builtin_amdgcn_s_wait_tensorcnt(2);
#else
    asm volatile("s_wait_tensorcnt 0x2" ::: "memory");
#endif
}
#endif // HAVE_TDM

// One 16-wide K-chunk of WMMAs for one wave (rows wave*16..+16, 64 cols).
__device__ __forceinline__ void mma_chunk(
    const float (&Asb)[TBM][TBK + 1], const float (&Bsb)[TBK][TBN + 4],
    v8f acc[4], int wave, int lane)
{
    const int mrow  = wave * 16 + (lane & 15);
    const int khalf = (lane >> 4) * 2;   // lanes 16-31 hold K+2,K+3
    #pragma unroll
    for (int ks = 0; ks < 4; ++ks) {
        const int kb = ks * 4 + khalf;
        v2f a;
        a.x = Asb[mrow][kb];
        a.y = Asb[mrow][kb + 1];
        #pragma unroll
        for (int ct = 0; ct < 4; ++ct) {
            const int col = ct * 16 + (lane & 15);
            v2f b;
            b.x = Bsb[kb][col];
            b.y = Bsb[kb + 1][col];
            acc[ct] = __builtin_amdgcn_wmma_f32_16x16x4_f32(
                false, a, false, b, (short)0, acc[ct], false, false);
        }
    }
}

// Generic WMMA GEMM:
//   C[out_map(r,c)] = alpha * sum_k A[r*a_rs + k*a_cs] * B[k*b_rs + c*b_cs]
//                     (+ I if addI && r==c)
// out addr = (r%rA)*rsA + ((r/rA)%rB)*rsB + (r/(rA*rB))*rsC
//          + (c%cA)*csA + ((c/cA)%cB)*csB + (c/(cA*cB))*csC
__global__ __launch_bounds__(128) void gemm_wmma_f32(
    const float* __restrict__ A, const float* __restrict__ B,
    float* __restrict__ C, int M, int N, int K,
    int a_rs, int a_cs, int b_rs, int b_cs,
    float alpha, int addI,
    int rA, int rsA, int rB, int rsB, int rsC,
    int cA, int csA, int cB, int csB, int csC)
{
    __shared__ float As[2][TBM][TBK + 1];   // 2 x 64x17
    __shared__ float Bs[2][TBK][TBN + 4];   // 2 x 16x68
    const int tid  = threadIdx.x;
    const int wave = tid >> 5;
    const int lane = tid & 31;
    const int bm = blockIdx.y * TBM;
    const int bn = blockIdx.x * TBN;
    const int nk = (K + TBK - 1) / TBK;

    v8f acc[4] = {};

#if HAVE_TDM
    if (a_cs == 1) {
        // --- TDM double-buffered path: both operands dim0-contiguous ---
        if (wave == 0) {
            tdm_tile_load((unsigned)(uintptr_t)&As[0][0][0],
                          A + (size_t)bm * a_rs, M - bm, K,
                          16u, 64u, (unsigned)a_rs, 3u, 0u);
            tdm_tile_load((unsigned)(uintptr_t)&Bs[0][0][0],
                          B + bn, K, N - bn,
                          64u, 16u, (unsigned)b_rs, 5u, 3u);
        }
        for (int it = 0; it < nk; ++it) {
            if (wave == 0) {
                if (it + 1 < nk) {
                    const int k1 = (it + 1) * TBK;
                    const int nb = (it + 1) & 1;
                    tdm_tile_load((unsigned)(uintptr_t)&As[nb][0][0],
                                  A + (size_t)bm * a_rs + k1, M - bm, K - k1,
                                  16u, 64u, (unsigned)a_rs, 3u, 0u);
                    tdm_tile_load((unsigned)(uintptr_t)&Bs[nb][0][0],
                                  B + (size_t)k1 * b_rs + bn, K - k1, N - bn,
                                  64u, 16u, (unsigned)b_rs, 5u, 3u);
                    tdm_wait2();   // current tiles done, prefetch in flight
                } else {
                    tdm_wait0();
                }
            }
            __syncthreads();
            mma_chunk(As[it & 1], Bs[it & 1], acc, wave, lane);
            __syncthreads();
        }
    } else
#endif
    {
        // --- manual single-buffered fallback (strided A, e.g. GEMM1) ---
        for (int k0 = 0; k0 < K; k0 += TBK) {
            #pragma unroll
            for (int t = 0; t < 8; ++t) {
                int idx = tid + t * 128;
                int m = idx >> 4, kk = idx & 15;
                int gm = bm + m, gk = k0 + kk;
                As[0][m][kk] = (gm < M && gk < K)
                    ? A[(size_t)gm * a_rs + (size_t)gk * a_cs] : 0.0f;
            }
            #pragma unroll
            for (int t = 0; t < 8; ++t) {
                int idx = tid + t * 128;
                int kk = idx >> 6, nn = idx & 63;
                int gk = k0 + kk, gn = bn + nn;
                Bs[0][kk][nn] = (gk < K && gn < N)
                    ? B[(size_t)gk * b_rs + (size_t)gn * b_cs] : 0.0f;
            }
            __syncthreads();
            mma_chunk(As[0], Bs[0], acc, wave, lane);
            __syncthreads();
        }
    }

    // Epilogue: C/D layout -> row = v + 8*(lane>=16), col = lane%16
    const int rbase = bm + wave * 16 + ((lane >> 4) << 3);
    const int cloc  = lane & 15;
    #pragma unroll
    for (int ct = 0; ct < 4; ++ct) {
        const int gc = bn + ct * 16 + cloc;
        if (gc >= N) continue;
        const long coff = (long)(gc % cA) * csA
                        + (long)((gc / cA) % cB) * csB
                        + (long)(gc / (cA * cB)) * csC;
        #pragma unroll
        for (int v = 0; v < 8; ++v) {
            const int gr = rbase + v;
            if (gr >= M) continue;
            const long roff = (long)(gr % rA) * rsA
                            + (long)((gr / rA) % rB) * rsB
                            + (long)(gr / (rA * rB)) * rsC;
            float val = alpha * acc[ct][v];
            if (addI && gr == gc) val += 1.0f;
            C[roff + coff] = val;
        }
    }
}

// ------------------------------ small kernels ------------------------------

__global__ void k_zero_f(float* p) { p[0] = 0.0f; }
__global__ void k_reset_u(unsigned* p) { p[0] = 0u; }

__global__ void k_copy(const float* __restrict__ s, float* __restrict__ d, int n) {
    int i = blockIdx.x * blockDim.x + threadIdx.x;
    if (i < n) d[i] = s[i];
}

// Antisymmetric A from triu segment (np.triu_indices order)
__global__ void k_build_A(const float* __restrict__ seg, float* __restrict__ Am, int n) {
    int i = blockIdx.x * blockDim.x + threadIdx.x;
    if (i >= n * n) return;
    int r = i / n, c = i % n;
    float v = 0.0f;
    if (r < c) { int p = r * n - r * (r + 1) / 2 + c - r - 1; v = seg[p]; }
    else if (r > c) { int p = c * n - c * (c + 1) / 2 + r - c - 1; v = -seg[p]; }
    Am[i] = v;
}

__global__ void k_scale(const float* __restrict__ a, float* __restrict__ b, int n, float s) {
    int i = blockIdx.x * blockDim.x + threadIdx.x;
    if (i < n) b[i] = a[i] * s;
}

__global__ void k_identity(float* __restrict__ e, int n) {
    int i = blockIdx.x * blockDim.x + threadIdx.x;
    if (i < n * n) e[i] = (i / n == i % n) ? 1.0f : 0.0f;
}

// U(d0,d0,dc) = E[:, :dc]  (row-major slice)
__global__ void k_slice_U(const float* __restrict__ E, float* __restrict__ U, int n, int dc) {
    int i = blockIdx.x * blockDim.x + threadIdx.x;
    if (i >= n * dc) return;
    int r = i / dc, a = i % dc;
    U[i] = E[(size_t)r * n + a];
}

__global__ void k_absmax(const float* __restrict__ z, unsigned* __restrict__ lam, int n) {
    int i = blockIdx.x * blockDim.x + threadIdx.x;
    if (i < n) atomicMax(lam, __float_as_uint(fabsf(z[i])));
}

// Z(b,e,a,c) -> T(b,c,e,a), scaled by 1/lam
__global__ void k_permute_scale(const float* __restrict__ Z, float* __restrict__ T,
                                const unsigned* __restrict__ lam,
                                int t1, int t3, int dc) {
    int i = blockIdx.x * blockDim.x + threadIdx.x;
    int total = t1 * t3 * dc * dc;
    if (i >= total) return;
    int c = i % dc;
    int a = (i / dc) % dc;
    int e = (i / (dc * dc)) % t3;
    int b = i / (dc * dc * t3);
    float inv = 1.0f / __uint_as_float(lam[0]);
    T[(((size_t)(b * dc + c) * t3 + e) * dc) + a] = Z[i] * inv;
}

__global__ void k_fe(float* __restrict__ fe, const unsigned* __restrict__ lam, float coef) {
    fe[0] -= coef * logf(__uint_as_float(lam[0]));
}

// ------------------------------ host driver --------------------------------

static inline void launch_gemm(hipStream_t st,
    const float* A, const float* B, float* C, int M, int N, int K,
    int a_rs, int a_cs, int b_rs, int b_cs, float alpha, int addI,
    int rA, int rsA, int rB, int rsB, int rsC,
    int cA, int csA, int cB, int csB, int csC)
{
    dim3 grid((N + TBN - 1) / TBN, (M + TBM - 1) / TBM), blk(128);
    gemm_wmma_f32<<<grid, blk, 0, st>>>(A, B, C, M, N, K,
        a_rs, a_cs, b_rs, b_cs, alpha, addI,
        rA, rsA, rB, rsB, rsC, cA, csA, cB, csB, csC);
}

static inline void launch_gemm_plain(hipStream_t st,
    const float* A, const float* B, float* C, int M, int N, int K,
    float alpha, int addI)
{
    launch_gemm(st, A, B, C, M, N, K, K, 1, N, 1, alpha, addI,
                M, N, 1, 0, 0, N, 1, 1, 0, 0);
}

extern "C" void kernel_launch(void* const* d_in, const int* in_sizes, int n_in,
                              void* d_out, int out_size, void* d_ws, size_t ws_size,
                              hipStream_t stream)
{
    const float* Aflat = (const float*)d_in[0];
    const float* T0    = (const float*)d_in[1];
    float* fe = (float*)d_out;

    // Workspace partition (floats)
    const size_t NMAX = 784;            // largest expm matrix dim
    const size_t NN   = NMAX * NMAX;    // 614656
    const size_t BIG  = 28UL * 28 * 28 * 28 * 28; // 17,210,368
    const size_t NEED = (6 * NN + 21952 + 2 * BIG + 4) * sizeof(float);
    if (ws_size < NEED) return;

    float* w   = (float*)d_ws;
    float* Am  = w;  w += NN;
    float* Sm  = w;  w += NN;
    float* E0  = w;  w += NN;
    float* E1  = w;  w += NN;
    float* Tb  = w;  w += NN;   // current T (max 28^4)
    float* Zb  = w;  w += NN;   // projection output
    float* Uf  = w;  w += 21952;
    float* X2  = w;  w += BIG;  // (b,m,a,j,n)
    float* Y2  = w;  w += BIG;  // (b,e,a,m,p)
    unsigned* lam = (unsigned*)w;

    const float TEMP = 2.269f;
    const int   KT   = 12;   // Taylor terms
    const int   SQ   = 8;    // squarings (scale A by 2^-8)

    k_zero_f<<<1, 1, 0, stream>>>(fe);
    k_copy<<<1, 32, 0, stream>>>(T0, Tb, 16);

    int t0 = 2, t1 = 2, t2 = 2, t3 = 2;
    long off = 0;

    for (int i = 0; i < 12; ++i) {
        const int n  = t0 * t0;
        const int dc = (t2 * t2 < 28) ? t2 * t2 : 28;
        const int d0 = t0;

        // ---- expm(A): scale + Taylor-Horner + squarings (all GEMMs) ----
        {
            int nn = n * n;
            k_build_A<<<(nn + 255) / 256, 256, 0, stream>>>(Aflat + off, Am, n);
            k_scale<<<(nn + 255) / 256, 256, 0, stream>>>(Am, Sm, nn, 1.0f / 256.0f);
            k_identity<<<(nn + 255) / 256, 256, 0, stream>>>(E0, n);
            float* cur = E0; float* nxt = E1;
            for (int k = KT; k >= 1; --k) {   // E = (S @ E)/k + I
                launch_gemm_plain(stream, Sm, cur, nxt, n, n, n, 1.0f / (float)k, 1);
                float* t = cur; cur = nxt; nxt = t;
            }
            for (int q = 0; q < SQ; ++q) {    // E = E @ E
                launch_gemm_plain(stream, cur, cur, nxt, n, n, n, 1.0f, 0);
                float* t = cur; cur = nxt; nxt = t;
            }
            k_slice_U<<<(n * dc + 255) / 256, 256, 0, stream>>>(cur, Uf, n, dc);
        }

        // ---- GEMM1: X[(j,a),(b,m,n)] = sum_i U[i,(j,a)] T[i,(b,m,n)]
        //      fused epilogue -> X2 layout (b,m,a,j,n)
        const int M1 = d0 * dc, N1 = t1 * t2 * t3, K1 = t0;
        const int K2 = t0 * t3;   // == t0*t1
        launch_gemm(stream, Uf, Tb, X2, M1, N1, K1,
                    /*A stored (K,M):*/ 1, M1, /*B row-major (K,N):*/ N1, 1,
                    1.0f, 0,
                    /*rows (j,a): a->K2, j->t3*/ dc, K2, t0, t3, 0,
                    /*cols (b,m,n): n->1, (b*t2+m)->dc*K2*/ t3, 1, t1 * t2, dc * K2, 0);

        // ---- GEMM2: Y[(b,m,a),(p,e)] = sum_{j,n} X2 * T[(j,n),(p,e)]
        //      fused epilogue -> Y2 layout (b,e,a,m,p)
        const int M2 = t1 * t2 * dc, N2 = t2 * t3;
        launch_gemm(stream, X2, Tb, Y2, M2, N2, K2,
                    K2, 1, N2, 1, 1.0f, 0,
                    /*rows (b,m,a): a->t2^2, m->t2, b->t3*dc*t2^2*/
                    dc, t2 * t2, t2, t2, t3 * dc * t2 * t2,
                    /*cols (p,e): e->dc*t2^2, p->1*/
                    t3, dc * t2 * t2, t2, 1, 0);

        // ---- GEMM3: Z[(b,e,a),c] = sum_{m,p} Y2 * U[(m,p),c]   (plain)
        const int M3 = t1 * t3 * dc, K3 = t2 * t2, N3 = dc;
        launch_gemm_plain(stream, Y2, Uf, Zb, M3, N3, K3, 1.0f, 0);

        // ---- lam = max|Z|, T <- permute(Z)/lam, fe update ----
        k_reset_u<<<1, 1, 0, stream>>>(lam);
        k_absmax<<<(M3 * N3 + 255) / 256, 256, 0, stream>>>(Zb, lam, M3 * N3);
        k_permute_scale<<<(M3 * N3 + 255) / 256, 256, 0, stream>>>(Zb, Tb, lam, t1, t3, dc);
        k_fe<<<1, 1, 0, stream>>>(fe, lam, TEMP / (float)(1 << (i + 1)));

        off += (long)n * (n - 1) / 2;
        const int nt0 = t1, nt1 = dc, nt2 = t3, nt3 = dc;
        t0 = nt0; t1 = nt1; t2 = nt2; t3 = nt3;
    }
}